// heteroRGCNLayer_70205535421296
// MI455X (gfx1250) — compile-verified
//
#include <hip/hip_runtime.h>
#include <hip/hip_bf16.h>
#include <cstddef>

typedef __attribute__((ext_vector_type(2))) float v2f;
typedef __attribute__((ext_vector_type(8))) float v8f;

#define DIM   128
#define NUSER 50000

// ---------------------------------------------------------------------------
// Wh[M,128] = feat[M,128] @ W[128,128] + b   (fp32 WMMA 16x16x4)
// grid.x = M/16 (row tiles); block = 256 = 8 waves; wave w handles col tile w.
// No control flow => EXEC is all ones for every WMMA, as the ISA requires.
// A layout (16x4 f32): lanes 0-15 carry M=lane, v[0]=K0,v[1]=K1;
//                      lanes 16-31 carry M=lane-16, v[0]=K2,v[1]=K3.
// B layout (4x16 f32): lanes 0-15 carry N=lane with K0/K1 in v[0]/v[1];
//                      lanes 16-31 carry K2/K3.
// C/D layout: VGPR r holds row (r + 8*(lane>=16)), col lane%16.
// ---------------------------------------------------------------------------
__global__ __launch_bounds__(256) void gemm_bias_wmma(
    const float* __restrict__ feat, const float* __restrict__ W,
    const float* __restrict__ bias, float* __restrict__ Wh) {
  const int lane = threadIdx.x & 31;
  const int wv   = threadIdx.x >> 5;      // 0..7 -> column tile
  const int row0 = blockIdx.x << 4;
  const int col0 = wv << 4;
  const int lm   = lane & 15;
  const int hi   = lane >> 4;             // selects K pair (0 -> K0/K1, 1 -> K2/K3)

  const float* aptr = feat + (size_t)(row0 + lm) * DIM + (hi << 1);
  const float* bptr = W    + (size_t)(hi << 1) * DIM + (col0 + lm);

  v8f c = {};
#pragma unroll
  for (int k = 0; k < DIM; k += 4) {
    v2f a = *(const v2f*)(aptr + k);      // 8B-aligned (even f32 index)
    v2f b;
    b.x = bptr[(size_t)k * DIM];
    b.y = bptr[(size_t)k * DIM + DIM];
    c = __builtin_amdgcn_wmma_f32_16x16x4_f32(
        /*neg_a=*/false, a, /*neg_b=*/false, b,
        /*c_mod=*/(short)0, c, /*reuse_a=*/false, /*reuse_b=*/false);
  }

  const float bv = bias[col0 + lm];
  float* out = Wh + (size_t)(row0 + (hi << 3)) * DIM + col0 + lm;
#pragma unroll
  for (int r = 0; r < 8; ++r)
    out[(size_t)r * DIM] = c[r] + bv;
}

// ---------------------------------------------------------------------------
// One wave per edge: lane l accumulates columns 4l..4l+3 of Wh[src] into
// s[dst] with f32 global atomics; lane 0 counts the edge for the mean.
// ---------------------------------------------------------------------------
__global__ __launch_bounds__(256) void scatter_mean_num(
    const float* __restrict__ Wh, const int* __restrict__ src,
    const int* __restrict__ dst, float* __restrict__ s,
    float* __restrict__ cnt, int E) {
  const int e    = (int)((blockIdx.x * 256u + threadIdx.x) >> 5);
  const int lane = threadIdx.x & 31;
  if (e >= E) return;
  const int si = src[e];
  const int di = dst[e];
  const float4 v = *((const float4*)(Wh + (size_t)si * DIM) + lane);
  float* o = s + (size_t)di * DIM + (lane << 2);
  atomicAdd(o + 0, v.x);
  atomicAdd(o + 1, v.y);
  atomicAdd(o + 2, v.z);
  atomicAdd(o + 3, v.w);
  if (lane == 0) atomicAdd(cnt + di, 1.0f);
}

__global__ __launch_bounds__(256) void zero_f32(float* __restrict__ p, int n) {
  const int i = blockIdx.x * 256 + threadIdx.x;
  if (i < n) p[i] = 0.0f;
}

// out = (cnt>0) ? s/cnt : 0, optionally accumulated (second etype).
__global__ __launch_bounds__(256) void finalize_mean(
    const float* __restrict__ s, const float* __restrict__ cnt,
    float* __restrict__ out, int accumulate) {
  const int i = blockIdx.x * 256 + threadIdx.x;   // exact grid: NUSER*DIM/256
  const int u = i >> 7;                           // i / 128
  const float c = cnt[u];
  const float m = (c > 0.0f) ? s[i] / c : 0.0f;
  if (accumulate) out[i] += m;
  else            out[i] = m;
}

extern "C" void kernel_launch(void* const* d_in, const int* in_sizes, int n_in,
                              void* d_out, int out_size, void* d_ws, size_t ws_size,
                              hipStream_t stream) {
  const float* feat  = (const float*)d_in[0];
  const float* Wc    = (const float*)d_in[1];
  const float* bc    = (const float*)d_in[2];
  const float* Wd    = (const float*)d_in[3];
  const float* bd    = (const float*)d_in[4];
  const int*   src_c = (const int*)d_in[5];
  const int*   dst_c = (const int*)d_in[6];
  const int*   src_d = (const int*)d_in[7];
  const int*   dst_d = (const int*)d_in[8];
  float* out = (float*)d_out;

  const int M = in_sizes[0] / DIM;   // 100000 news nodes (multiple of 16)
  const int E = in_sizes[5];         // 800000 edges per etype

  // Workspace: Wh [M,128] | s [NUSER,128] | cnt [NUSER]  (~77 MB total)
  float* Wh  = (float*)d_ws;
  float* s   = Wh + (size_t)M * DIM;
  float* cnt = s + (size_t)NUSER * DIM;

  const int nzero = NUSER * DIM + NUSER;       // s and cnt are contiguous
  dim3 zgrid((nzero + 255) / 256);
  dim3 ggrid(M / 16);
  dim3 sgrid((E * 32 + 255) / 256);
  dim3 fgrid((NUSER * DIM) / 256);

  // ---- etype "clicked": mean into out ----
  zero_f32<<<zgrid, 256, 0, stream>>>(s, nzero);
  gemm_bias_wmma<<<ggrid, 256, 0, stream>>>(feat, Wc, bc, Wh);
  scatter_mean_num<<<sgrid, 256, 0, stream>>>(Wh, src_c, dst_c, s, cnt, E);
  finalize_mean<<<fgrid, 256, 0, stream>>>(s, cnt, out, 0);

  // ---- etype "disliked": mean added on top ----
  zero_f32<<<zgrid, 256, 0, stream>>>(s, nzero);
  gemm_bias_wmma<<<ggrid, 256, 0, stream>>>(feat, Wd, bd, Wh);
  scatter_mean_num<<<sgrid, 256, 0, stream>>>(Wh, src_d, dst_d, s, cnt, E);
  finalize_mean<<<fgrid, 256, 0, stream>>>(s, cnt, out, 1);
}